// Decoder_5823975654080
// MI455X (gfx1250) — compile-verified
//
#include <hip/hip_runtime.h>
#include <stdint.h>
#include <math.h>

// Problem constants (from reference)
#define V_  32000
#define E_  256
#define H_  512
#define B_  64
#define S_  64

#define KC_    128   // k-chunk staged in LDS per round (4 rounds over H=512)
#define PITCH_ 132   // padded LDS row pitch (floats): lanes hit banks 4 apart

typedef float v2f __attribute__((ext_vector_type(2)));
typedef float v4f __attribute__((ext_vector_type(4)));
typedef float v8f __attribute__((ext_vector_type(8)));
typedef int   v4i __attribute__((vector_size(16)));  // matches builtin param

#if __has_builtin(__builtin_amdgcn_global_load_async_to_lds_b128) && \
    __has_builtin(__builtin_amdgcn_s_wait_asynccnt)
#define USE_ASYNC_LDS 1
#else
#define USE_ASYNC_LDS 0
#endif

// D = A(16x4 f32) * B(4x16 f32) + C(16x16 f32), full fp32 precision.
__device__ __forceinline__ v8f wmma4(v2f a, v2f b, v8f c) {
  return __builtin_amdgcn_wmma_f32_16x16x4_f32(false, a, false, b, (short)0, c,
                                               false, false);
}

__device__ __forceinline__ float sigmoidf_(float x) {
  return 1.0f / (1.0f + expf(-x));
}

// Monotone mapping float -> uint32 so unsigned compare == float compare.
__device__ __forceinline__ unsigned int fkey(float x) {
  unsigned int u = __float_as_uint(x);
  return (u & 0x80000000u) ? ~u : (u | 0x80000000u);
}

// Stage h[:, kc:kc+128] into a padded LDS buffer (64 x PITCH_ floats).
// 256 threads x 8 float4 chunks = 64*128 floats.  Async on CDNA5.
__device__ __forceinline__ void stage_chunk(const float* __restrict__ h,
                                            int kc, float* buf, int tid) {
#pragma unroll
  for (int j = 0; j < 8; ++j) {
    const int f   = tid + 256 * j;  // float4 index, 0..2047
    const int row = f >> 5;         // 0..63
    const int c4  = f & 31;         // 0..31
    const float* g = h + (size_t)row * H_ + kc + c4 * 4;
    float* l = buf + row * PITCH_ + c4 * 4;
#if USE_ASYNC_LDS
    __builtin_amdgcn_global_load_async_to_lds_b128(
        (__attribute__((address_space(1))) v4i*)g,
        (__attribute__((address_space(3))) v4i*)l, 0, 0);
#else
    *(v4f*)l = *(const v4f*)g;
#endif
  }
}

// ---------------------------------------------------------------------------
// Init: h = encoder_hidden, c = 0, tok = captions[:,0], argmax = 0, zrow = 0
// ---------------------------------------------------------------------------
__global__ void k_init(const float* __restrict__ enc,
                       const int* __restrict__ captions,
                       float* __restrict__ h, float* __restrict__ c,
                       int* __restrict__ tok,
                       unsigned long long* __restrict__ amax,
                       float* __restrict__ zrow) {
  int i = blockIdx.x * blockDim.x + threadIdx.x;
  if (i < B_ * H_) { h[i] = enc[i]; c[i] = 0.0f; }
  if (i < E_) zrow[i] = 0.0f;
  if (i < B_) { tok[i] = captions[i * S_]; amax[i] = 0ull; }
}

// out[:, 0, :] = 0
__global__ void k_zero0(float* __restrict__ out) {
  size_t i = (size_t)blockIdx.x * blockDim.x + threadIdx.x;
  if (i < (size_t)B_ * V_) {
    size_t m = i / V_, n = i % V_;
    __builtin_nontemporal_store(0.0f, out + m * ((size_t)S_ * V_) + n);
  }
}

// ---------------------------------------------------------------------------
// LSTM cell step: gates = emb[tok] @ W_ih^T + h_in @ W_hh^T + (b_ih+b_hh),
// then c/h update.  Block = (16 batch rows) x (16 hidden cols) slab.
// 4 waves: wave g computes the 16x16 tile of gate g (i,f,g,o) with f32 WMMA.
// ---------------------------------------------------------------------------
__global__ __launch_bounds__(128) void k_cell(
    const float* __restrict__ emb, const float* __restrict__ Wih,
    const float* __restrict__ Whh, const float* __restrict__ bih,
    const float* __restrict__ bhh, const int* __restrict__ tok,
    const float* __restrict__ zrow, const float* __restrict__ h_in,
    float* __restrict__ h_out, float* __restrict__ c) {
  __shared__ float gtile[4][16][16];

  const int tid  = threadIdx.x;
  const int wave = tid >> 5;        // gate index (wave32)
  const int lane = tid & 31;
  const int l16  = lane & 15;
  const int hl   = lane >> 4;
  const int m0   = blockIdx.y * 16; // batch tile base
  const int n0   = blockIdx.x * 16; // hidden-col tile base
  const int gcol = wave * H_ + n0 + l16;  // column in [0,4H)

  // A-fragment row for this lane (lanes 0-15 and 16-31 both cover M=0..15)
  const int M    = m0 + l16;
  const int tokm = tok[M];
  const float* erow = (tokm != 0) ? emb + (size_t)tokm * E_ : zrow;  // row0 = 0
  const float* hrow = h_in + (size_t)M * H_;
  const float* wih  = Wih + (size_t)gcol * E_;
  const float* whh  = Whh + (size_t)gcol * H_;

  v8f acc = {};
  // K = 0..E-1 : embedding part
  for (int kk = 0; kk < E_; kk += 4) {
    const int K0 = kk + 2 * hl;               // this lane's K pair
    const v2f a = *(const v2f*)(erow + K0);
    const v2f b = *(const v2f*)(wih + K0);    // B[k][n] = W_ih[n][k]
    acc = wmma4(a, b, acc);
  }
  // K = E..E+H-1 : recurrent part
  for (int kk = 0; kk < H_; kk += 4) {
    const int K0 = kk + 2 * hl;
    const v2f a = *(const v2f*)(hrow + K0);
    const v2f b = *(const v2f*)(whh + K0);
    acc = wmma4(a, b, acc);
  }

  const float bias = bih[gcol] + bhh[gcol];
#pragma unroll
  for (int r = 0; r < 8; ++r)                  // D row = r + 8*hl, col = l16
    gtile[wave][r + 8 * hl][l16] = acc[r] + bias;
  __syncthreads();

  // Elementwise LSTM update over the 16x16 slab (2 elements / thread)
#pragma unroll
  for (int e = tid; e < 256; e += 128) {
    const int row = e >> 4, col = e & 15;
    const float iv = gtile[0][row][col];
    const float fv = gtile[1][row][col];
    const float gv = gtile[2][row][col];
    const float ov = gtile[3][row][col];
    const size_t idx = (size_t)(m0 + row) * H_ + (n0 + col);
    const float cp = c[idx];
    const float cn = sigmoidf_(fv) * cp + sigmoidf_(iv) * tanhf(gv);
    const float hn = sigmoidf_(ov) * tanhf(cn);
    c[idx]     = cn;
    h_out[idx] = hn;
  }
}

// ---------------------------------------------------------------------------
// logits = h @ fc_w^T + fc_b ; streamed to out[:, t, :] (non-temporal) and
// fused row-argmax via packed-u64 atomicMax.
// Block = 256 threads = 8 waves; each wave owns one 16-col tile of V and
// computes all 4 batch tiles.  The shared A-matrix (h) is staged once per
// block into double-buffered LDS via async-to-LDS copies (dedupes the 8x
// redundant global reads and moves A onto the DS pipe).
// ---------------------------------------------------------------------------
__global__ __launch_bounds__(256) void k_logits(
    const float* __restrict__ h, const float* __restrict__ fcw,
    const float* __restrict__ fcb, float* __restrict__ out,
    unsigned long long* __restrict__ amax, int t) {
  __shared__ float hs[2][B_ * PITCH_];  // 2 x 33 KB

  const int tid  = threadIdx.x;
  const int wave = tid >> 5;
  const int lane = tid & 31;
  const int l16  = lane & 15;
  const int hl   = lane >> 4;
  const int n0   = (blockIdx.x * 8 + wave) * 16;  // vocab tile base
  const int n    = n0 + l16;
  const float* wrow = fcw + (size_t)n * H_;       // B[k][n] = fc_w[n][k]

  stage_chunk(h, 0, hs[0], tid);  // prologue: chunk 0 in flight

  v8f acc[4] = {};
  int buf = 0;
#pragma unroll
  for (int ch = 0; ch < 4; ++ch) {
    if (ch < 3) stage_chunk(h, (ch + 1) * KC_, hs[buf ^ 1], tid);  // prefetch
#if USE_ASYNC_LDS
    if (ch < 3) __builtin_amdgcn_s_wait_asynccnt(8);  // oldest 8 done
    else        __builtin_amdgcn_s_wait_asynccnt(0);
#endif
    __syncthreads();  // all waves' pieces of chunk `ch` visible

    const float* A = hs[buf];
    for (int kk = 0; kk < KC_; kk += 4) {
      const int K0 = kk + 2 * hl;
      const v2f b = *(const v2f*)(wrow + ch * KC_ + K0);  // global (L2)
#pragma unroll
      for (int mt = 0; mt < 4; ++mt) {                    // LDS, conflict-free
        const v2f a = *(const v2f*)(A + (mt * 16 + l16) * PITCH_ + K0);
        acc[mt] = wmma4(a, b, acc[mt]);
      }
    }
    __syncthreads();  // done reading `buf` before it is refilled
    buf ^= 1;
  }

  const float bias = fcb[n];
#pragma unroll
  for (int mt = 0; mt < 4; ++mt) {
#pragma unroll
    for (int r = 0; r < 8; ++r) {
      const float v = acc[mt][r] + bias;
      const int m = mt * 16 + r + 8 * hl;
      // stream logits; keep L2 for fc_w
      __builtin_nontemporal_store(
          v, out + (size_t)m * ((size_t)S_ * V_) + (size_t)t * V_ + n);
      // argmax key: bigger packed == bigger logit, ties -> lower index
      unsigned long long p =
          ((unsigned long long)fkey(v) << 32) |
          (unsigned long long)(0xFFFFFFFFu - (unsigned)n);
      // reduce over the 16 lanes holding the same row m
#pragma unroll
      for (int off = 1; off < 16; off <<= 1) {
        const unsigned lo = __shfl_xor((unsigned)(p & 0xFFFFFFFFu), off, 32);
        const unsigned hi = __shfl_xor((unsigned)(p >> 32), off, 32);
        const unsigned long long q = ((unsigned long long)hi << 32) | lo;
        if (q > p) p = q;
      }
      if (l16 == 0) atomicMax(&amax[m], p);
    }
  }
}

// ---------------------------------------------------------------------------
// Next token: teacher forcing or argmax; reset argmax slots for next step.
// ---------------------------------------------------------------------------
__global__ void k_next(unsigned long long* __restrict__ amax,
                       const int* __restrict__ captions,
                       const unsigned char* __restrict__ use_tf,
                       int* __restrict__ tok, int t) {
  const int m = threadIdx.x;
  if (m < B_) {
    const unsigned long long p = amax[m];
    const int n = (int)(0xFFFFFFFFu - (unsigned)(p & 0xFFFFFFFFu));
    tok[m] = use_tf[t] ? captions[m * S_ + t] : n;
    amax[m] = 0ull;
  }
}

// ---------------------------------------------------------------------------
extern "C" void kernel_launch(void* const* d_in, const int* in_sizes, int n_in,
                              void* d_out, int out_size, void* d_ws,
                              size_t ws_size, hipStream_t stream) {
  const float* enc      = (const float*)d_in[0];
  const int*   captions = (const int*)d_in[1];
  const unsigned char* use_tf = (const unsigned char*)d_in[2];  // jnp bool = 1B
  const float* emb = (const float*)d_in[3];
  const float* Wih = (const float*)d_in[4];
  const float* Whh = (const float*)d_in[5];
  const float* bih = (const float*)d_in[6];
  const float* bhh = (const float*)d_in[7];
  const float* fcw = (const float*)d_in[8];
  const float* fcb = (const float*)d_in[9];
  float* out = (float*)d_out;

  // workspace: hA | hB | c | zrow(256f) | amax(u64 x 64) | tok(int x 64)
  float* hA   = (float*)d_ws;
  float* hB   = hA + B_ * H_;
  float* c    = hB + B_ * H_;
  float* zrow = c + B_ * H_;
  unsigned long long* amax = (unsigned long long*)(zrow + E_);
  int* tok = (int*)(amax + B_);

  k_init<<<(B_ * H_ + 255) / 256, 256, 0, stream>>>(enc, captions, hA, c, tok,
                                                    amax, zrow);
  k_zero0<<<(B_ * V_ + 255) / 256, 256, 0, stream>>>(out);

  for (int t = 1; t < S_; ++t) {
    float* h_in  = (t & 1) ? hA : hB;   // double-buffer h across steps
    float* h_out = (t & 1) ? hB : hA;
    k_cell<<<dim3(H_ / 16, B_ / 16), 128, 0, stream>>>(
        emb, Wih, Whh, bih, bhh, tok, zrow, h_in, h_out, c);
    k_logits<<<dim3(V_ / 128), 256, 0, stream>>>(h_out, fcw, fcb, out, amax, t);
    k_next<<<1, 64, 0, stream>>>(amax, captions, use_tf, tok, t);
  }
}